// Augmentation3d_1434519077377
// MI455X (gfx1250) — compile-verified
//
#include <hip/hip_runtime.h>
#include <math.h>

typedef __attribute__((ext_vector_type(2))) float v2f;
typedef __attribute__((ext_vector_type(8))) float v8f;

// ---- geometry ----
#define VOL_D   160
#define OUT_D   128
#define OFFS    16              // (160-128)/2
#define KN      15              // raw knot grid per axis
#define B1_F    (160*15*15)     // 36000 floats per field
#define B2_F    (160*160*15)    // 384000 floats per field

// ---- workspace layout (float offsets) ----
#define WS_A     0              // [160*15]  combined resize∘crop∘S^4 matrix
#define WS_THETA 2400           // [4*12]    per-batch 3x4 affine
#define WS_B1    2448           // [3][B1_F]
#define WS_B2    110448         // [3][B2_F]   (total ~4.8 MB)

// =====================================================================
// Kernel 1: build constant per-axis operator A (160x15) and theta (4x3x4)
// =====================================================================
__global__ void prep_kernel(const float* __restrict__ rot,
                            const float* __restrict__ scale,
                            const float* __restrict__ shift,
                            float* __restrict__ ws) {
  __shared__ float Ssm[225], S2sm[225], S4sm[225];
  int tid = threadIdx.x;
  // S[i][j]: 5-tap box blur with edge clamp on a 15-vector
  if (tid < 225) {
    int i = tid / 15, j = tid % 15;
    int cnt = 0;
    for (int t = -2; t <= 2; ++t) {
      int jj = i + t; jj = jj < 0 ? 0 : (jj > 14 ? 14 : jj);
      cnt += (jj == j) ? 1 : 0;
    }
    Ssm[tid] = 0.2f * (float)cnt;
  }
  __syncthreads();
  if (tid < 225) {              // S2 = S*S
    int i = tid / 15, j = tid % 15;
    float acc = 0.f;
    for (int k = 0; k < 15; ++k) acc += Ssm[i*15+k] * Ssm[k*15+j];
    S2sm[tid] = acc;
  }
  __syncthreads();
  if (tid < 225) {              // S4 = S2*S2
    int i = tid / 15, j = tid % 15;
    float acc = 0.f;
    for (int k = 0; k < 15; ++k) acc += S2sm[i*15+k] * S2sm[k*15+j];
    S4sm[tid] = acc;
  }
  __syncthreads();
  // A[o][i] = resize(13->160) of rows 2..14 of S4  (linear interp, half-pixel)
  float* A = ws + WS_A;
  for (int e = tid; e < 160*15; e += blockDim.x) {
    int o = e / 15, i = e % 15;
    float src = ((float)o + 0.5f) * (13.0f/160.0f) - 0.5f;
    src = fminf(fmaxf(src, 0.0f), 12.0f);
    int i0 = (int)floorf(src);
    int i1 = i0 + 1; if (i1 > 12) i1 = 12;
    float w = src - (float)i0;
    A[e] = (1.0f - w) * S4sm[(2+i0)*15 + i] + w * S4sm[(2+i1)*15 + i];
  }
  // Rodrigues -> theta = [R*scale | shift]
  if (tid < 4) {
    int b = tid;
    float rx = rot[b*3+0], ry = rot[b*3+1], rz = rot[b*3+2];
    const float eps = 1e-6f;
    float t2 = rx*rx + ry*ry + rz*rz;
    float th = sqrtf(fmaxf(t2, eps));
    float wxn = rx/(th+eps), wyn = ry/(th+eps), wzn = rz/(th+eps);
    float c = cosf(th), s = sinf(th), k = 1.0f - c;
    float R[9];
    if (t2 > eps) {
      R[0]=c+wxn*wxn*k;       R[1]=wxn*wyn*k-wzn*s;   R[2]=wyn*s+wxn*wzn*k;
      R[3]=wzn*s+wxn*wyn*k;   R[4]=c+wyn*wyn*k;       R[5]=-wxn*s+wyn*wzn*k;
      R[6]=-wyn*s+wxn*wzn*k;  R[7]=wxn*s+wyn*wzn*k;   R[8]=c+wzn*wzn*k;
    } else {
      R[0]=1.f;  R[1]=-rz;  R[2]=ry;
      R[3]=rz;   R[4]=1.f;  R[5]=-rx;
      R[6]=-ry;  R[7]=rx;   R[8]=1.f;
    }
    float* to = ws + WS_THETA + b*12;
    for (int i = 0; i < 3; ++i) {
      for (int j = 0; j < 3; ++j) to[i*4+j] = R[i*3+j] * scale[b*3+j];
      to[i*4+3] = shift[b*3+i];
    }
  }
}

// =====================================================================
// One wave computes one 16x16 tile of C = A(Mx15) * (bScale*B + bBias)(15xN)
// via 4x V_WMMA_F32_16X16X4_F32.  K padded 15->16 with masked operands.
// Masking is select-based (no branches) so EXEC stays all-ones.
// =====================================================================
__device__ __forceinline__ void wmma_tile(const float* __restrict__ A,
                                          const float* __restrict__ B,
                                          float* __restrict__ C,
                                          int M, int N, int tileM, int tileN,
                                          float bScale, float bBias) {
  int lane = threadIdx.x & 31;
  int half = lane >> 4;        // 0: lanes 0-15, 1: lanes 16-31
  int l16  = lane & 15;
  int m = tileM*16 + l16;
  int n = tileN*16 + l16;
  int mc = m < M ? m : M-1;
  int nc = n < N ? n : N-1;
  float mMask = m < M ? 1.f : 0.f;
  float nMask = n < N ? 1.f : 0.f;
  v8f acc = {0.f,0.f,0.f,0.f,0.f,0.f,0.f,0.f};
  #pragma unroll
  for (int kc = 0; kc < 4; ++kc) {
    v2f av, bv;
    #pragma unroll
    for (int v = 0; v < 2; ++v) {
      int kk  = kc*4 + half*2 + v;          // ISA A/B 32-bit layout: K = 2*(l/16)+v
      int kkc = kk < 15 ? kk : 14;
      float kMask = kk < 15 ? 1.f : 0.f;
      av[v] = A[mc*15 + kkc] * (mMask * kMask);
      bv[v] = (B[kkc*N + nc] * bScale + bBias) * (nMask * kMask);
    }
    acc = __builtin_amdgcn_wmma_f32_16x16x4_f32(
        /*neg_a=*/false, av, /*neg_b=*/false, bv,
        /*c_mod=*/(short)0, acc, /*reuse_a=*/false, /*reuse_b=*/false);
  }
  #pragma unroll
  for (int r = 0; r < 8; ++r) {             // C/D layout: row = r + 8*(l/16)
    int mm = tileM*16 + r + 8*half;
    int nn = tileN*16 + l16;
    if (mm < M && nn < N) C[mm*N + nn] = acc[r];
  }
}

// Stage 1: B1_f[z, (j,k)] = sum_i A[z,i] * (4*d_f[i,j,k] - 2)   (M=160, N=225)
__global__ void field_stage1(const float* __restrict__ dz,
                             const float* __restrict__ dy,
                             const float* __restrict__ dx,
                             float* __restrict__ ws) {
  const float* A = ws + WS_A;
  const float* d = (blockIdx.z == 0) ? dz : ((blockIdx.z == 1) ? dy : dx);
  float* C = ws + WS_B1 + (size_t)blockIdx.z * B1_F;
  wmma_tile(A, d, C, 160, 225, blockIdx.x, blockIdx.y, 4.0f, -2.0f);
}

// Stage 2: B2_f[z, y, k] = sum_j A[y,j] * B1_f[z, j, k]   (per z: M=160, N=15)
__global__ void field_stage2(float* __restrict__ ws) {
  int f = blockIdx.z / 160;
  int z = blockIdx.z % 160;
  const float* A = ws + WS_A;
  const float* B = ws + WS_B1 + (size_t)f * B1_F + z * 225;
  float* C       = ws + WS_B2 + (size_t)f * B2_F + z * 2400;
  wmma_tile(A, B, C, 160, 15, blockIdx.x, 0, 1.0f, 0.0f);
}

// =====================================================================
// Stage 3 fused with affine grid + trilinear sample of the cropped 128^3.
// disp_f(z,y,x) = sum_k A[x,k] * B2_f[z,y,k]  (B2 row uniform per wave)
// =====================================================================
__global__ __launch_bounds__(128) void sample_kernel(
    const float* __restrict__ vol, const float* __restrict__ ws,
    float* __restrict__ out) {
  int xo = threadIdx.x, yo = blockIdx.x, zo = blockIdx.y, b = blockIdx.z;
  int x = xo + OFFS, y = yo + OFFS, z = zo + OFFS;
  const float inv = 1.0f / 160.0f;
  float xs = (2.0f*(float)x + 1.0f)*inv - 1.0f;
  float ys = (2.0f*(float)y + 1.0f)*inv - 1.0f;
  float zs = (2.0f*(float)z + 1.0f)*inv - 1.0f;

  const float* Arow = ws + WS_A + x*15;                 // per-lane (15 floats)
  const float* B2r  = ws + WS_B2 + ((size_t)z*160 + y)*15;  // wave-uniform
  float d0 = 0.f, d1 = 0.f, d2 = 0.f;
  #pragma unroll
  for (int k = 0; k < 15; ++k) {
    float a = Arow[k];
    d0 = fmaf(a, B2r[k],            d0);   // field(dz) -> grid x component
    d1 = fmaf(a, B2r[B1_F ? (B2_F + k) : k], d1);
    d2 = fmaf(a, B2r[2*B2_F + k],   d2);
  }
  const float* th = ws + WS_THETA + b*12;
  float gx = th[0]*xs + th[1]*ys + th[2]*zs  + th[3]  + d0;
  float gy = th[4]*xs + th[5]*ys + th[6]*zs  + th[7]  + d1;
  float gz = th[8]*xs + th[9]*ys + th[10]*zs + th[11] + d2;

  float fx = ((gx + 1.0f)*160.0f - 1.0f)*0.5f;
  float fy = ((gy + 1.0f)*160.0f - 1.0f)*0.5f;
  float fz = ((gz + 1.0f)*160.0f - 1.0f)*0.5f;
  float x0f = floorf(fx), y0f = floorf(fy), z0f = floorf(fz);
  int x0 = (int)x0f, y0 = (int)y0f, z0 = (int)z0f;
  float wx = fx - x0f, wy = fy - y0f, wz = fz - z0f;

  const float* vb = vol + (size_t)b * 160*160*160;
  auto tap = [&](int zi, int yi, int xi) -> float {
    bool ok = ((unsigned)zi < 160u) && ((unsigned)yi < 160u) && ((unsigned)xi < 160u);
    int zc = zi < 0 ? 0 : (zi > 159 ? 159 : zi);
    int yc = yi < 0 ? 0 : (yi > 159 ? 159 : yi);
    int xc = xi < 0 ? 0 : (xi > 159 ? 159 : xi);
    float v = vb[((size_t)zc*160 + yc)*160 + xc];      // unconditional load
    return ok ? v : 0.0f;                              // zero-pad outside
  };
  float r =
      tap(z0,   y0,   x0  ) * ((1.f-wz)*(1.f-wy)*(1.f-wx)) +
      tap(z0,   y0,   x0+1) * ((1.f-wz)*(1.f-wy)*wx)       +
      tap(z0,   y0+1, x0  ) * ((1.f-wz)*wy*(1.f-wx))       +
      tap(z0,   y0+1, x0+1) * ((1.f-wz)*wy*wx)             +
      tap(z0+1, y0,   x0  ) * (wz*(1.f-wy)*(1.f-wx))       +
      tap(z0+1, y0,   x0+1) * (wz*(1.f-wy)*wx)             +
      tap(z0+1, y0+1, x0  ) * (wz*wy*(1.f-wx))             +
      tap(z0+1, y0+1, x0+1) * (wz*wy*wx);

  out[(((size_t)b*128 + zo)*128 + yo)*128 + xo] = r;
}

extern "C" void kernel_launch(void* const* d_in, const int* in_sizes, int n_in,
                              void* d_out, int out_size, void* d_ws, size_t ws_size,
                              hipStream_t stream) {
  (void)in_sizes; (void)n_in; (void)out_size; (void)ws_size;
  const float* vol   = (const float*)d_in[0];
  const float* rot   = (const float*)d_in[1];
  const float* scale = (const float*)d_in[2];
  const float* shift = (const float*)d_in[3];
  const float* dz    = (const float*)d_in[4];
  const float* dy    = (const float*)d_in[5];
  const float* dx    = (const float*)d_in[6];
  float* out = (float*)d_out;
  float* ws  = (float*)d_ws;

  prep_kernel  <<<dim3(1),           256, 0, stream>>>(rot, scale, shift, ws);
  field_stage1 <<<dim3(10, 15, 3),    32, 0, stream>>>(dz, dy, dx, ws);
  field_stage2 <<<dim3(10, 1, 480),   32, 0, stream>>>(ws);
  sample_kernel<<<dim3(128, 128, 4), 128, 0, stream>>>(vol, ws, out);
}